// GNN_1l_73538430042257
// MI455X (gfx1250) — compile-verified
//
#include <hip/hip_runtime.h>

typedef __attribute__((ext_vector_type(2))) float v2f;
typedef __attribute__((ext_vector_type(8))) float v8f;

#define DHID  128
#define KPAIRS (DHID / 2)     // 64 K-pairs
#define NPAD  144             // float2 row stride: 288 dwords == 32 mod 64 banks
                              // -> half-waves hit disjoint LDS bank sets

// Stage W[128x128] into LDS in K-paired layout:
//   Wp[kp*NPAD + n] = { W[2kp][n], W[2kp+1][n] }
// so one B fragment (K=ka, ka+1 at col n) is a single ds_load_b64.
__device__ __forceinline__ void stage_W(const float* __restrict__ W, v2f* Wp)
{
    for (int i = threadIdx.x; i < KPAIRS * DHID; i += blockDim.x) {
        const int kp = i >> 7;         // / DHID
        const int n  = i & (DHID - 1);
        v2f p;
        p.x = W[(size_t)(2 * kp)     * DHID + n];
        p.y = W[(size_t)(2 * kp + 1) * DHID + n];
        Wp[(size_t)kp * NPAD + n] = p;
    }
}

// 16-row strip x 128 cols of  acc += A_strip @ W  using v_wmma_f32_16x16x4_f32.
// scale multiplies the A fragments (used to fold 1/deg into the mean path).
__device__ __forceinline__ void strip_gemm_accum(
    const float* __restrict__ arow,   // A + rowA*DHID  (rowA = row0 + (lane&15))
    const v2f* __restrict__ Wp,       // staged weights in LDS
    float scale, int half, int l16, v8f acc[8])
{
    for (int k0 = 0; k0 < DHID; k0 += 4) {
        const int ka = k0 + half * 2;
        v2f a = *(const v2f*)(arow + ka);
        a.x *= scale;
        a.y *= scale;
        const int kp = ka >> 1;
#pragma unroll
        for (int t = 0; t < 8; ++t) {
            const v2f b = Wp[(size_t)kp * NPAD + t * 16 + l16];
            acc[t] = __builtin_amdgcn_wmma_f32_16x16x4_f32(
                false, a, false, b, (short)0, acc[t], false, false);
        }
    }
}

// ---------------------------------------------------------------------------
// C[M x 128] = act(A[M x 128] @ W + b)
// One workgroup stages W once, then grid-strides waves over 16-row strips.
// ---------------------------------------------------------------------------
template <bool ACT>
__global__ void fc_kernel(const float* __restrict__ A,
                          const float* __restrict__ W,
                          const float* __restrict__ bias,
                          float* __restrict__ C, int M)
{
    extern __shared__ char smem_raw[];
    v2f* Wp = (v2f*)smem_raw;

    stage_W(W, Wp);
    __syncthreads();

    const int lane = threadIdx.x & 31;
    const int half = lane >> 4;
    const int l16  = lane & 15;
    const int wavesPerBlock = blockDim.x >> 5;
    const int waveId = threadIdx.x >> 5;
    const int stripStride = gridDim.x * wavesPerBlock;

    // bias per (tile, lane) is strip-invariant: hoist out of the loop
    float bv[8];
#pragma unroll
    for (int t = 0; t < 8; ++t) bv[t] = bias[t * 16 + l16];

    for (int strip = blockIdx.x * wavesPerBlock + waveId;
         strip * 16 < M; strip += stripStride) {
        const int row0 = strip * 16;
        v8f acc[8];
#pragma unroll
        for (int t = 0; t < 8; ++t) acc[t] = v8f{0, 0, 0, 0, 0, 0, 0, 0};

        strip_gemm_accum(A + (size_t)(row0 + l16) * DHID, Wp, 1.0f, half, l16, acc);

#pragma unroll
        for (int t = 0; t < 8; ++t) {
#pragma unroll
            for (int v = 0; v < 8; ++v) {
                const int row = row0 + v + half * 8;
                float val = acc[t][v] + bv[t];
                if (ACT) val = val > 0.0f ? val : 0.0f;
                C[(size_t)row * DHID + t * 16 + l16] = val;
            }
        }
    }
}

// ---------------------------------------------------------------------------
// SAGE combine:
//   C = act(Hself @ Wself + bself + (Agg/max(Deg,1)) @ Wneigh + bneigh)
// Both weight matrices staged in LDS (144 KB of the 320 KB/WGP pool);
// per-row 1/deg folded into the A fragments of the neighbor GEMM.
// ---------------------------------------------------------------------------
template <bool ACT>
__global__ void sage_kernel(const float* __restrict__ Hself,
                            const float* __restrict__ Agg,
                            const float* __restrict__ Deg,
                            const float* __restrict__ Wself,
                            const float* __restrict__ bself,
                            const float* __restrict__ Wneigh,
                            const float* __restrict__ bneigh,
                            float* __restrict__ C, int M)
{
    extern __shared__ char smem_raw[];
    v2f* WpS = (v2f*)smem_raw;
    v2f* WpN = WpS + (size_t)KPAIRS * NPAD;

    stage_W(Wself, WpS);
    stage_W(Wneigh, WpN);
    __syncthreads();

    const int lane = threadIdx.x & 31;
    const int half = lane >> 4;
    const int l16  = lane & 15;
    const int wavesPerBlock = blockDim.x >> 5;
    const int waveId = threadIdx.x >> 5;
    const int stripStride = gridDim.x * wavesPerBlock;

    float bv[8];
#pragma unroll
    for (int t = 0; t < 8; ++t)
        bv[t] = bself[t * 16 + l16] + bneigh[t * 16 + l16];

    for (int strip = blockIdx.x * wavesPerBlock + waveId;
         strip * 16 < M; strip += stripStride) {
        const int row0 = strip * 16;
        const int rowA = row0 + l16;          // fixed A row per lane

        v8f acc[8];
#pragma unroll
        for (int t = 0; t < 8; ++t) acc[t] = v8f{0, 0, 0, 0, 0, 0, 0, 0};

        // self path
        strip_gemm_accum(Hself + (size_t)rowA * DHID, WpS, 1.0f, half, l16, acc);

        // neighbor (mean) path: scale A by 1/max(deg,1)
        const float invd = 1.0f / fmaxf(Deg[rowA], 1.0f);
        strip_gemm_accum(Agg + (size_t)rowA * DHID, WpN, invd, half, l16, acc);

#pragma unroll
        for (int t = 0; t < 8; ++t) {
#pragma unroll
            for (int v = 0; v < 8; ++v) {
                const int row = row0 + v + half * 8;
                float val = acc[t][v] + bv[t];
                if (ACT) val = val > 0.0f ? val : 0.0f;
                C[(size_t)row * DHID + t * 16 + l16] = val;
            }
        }
    }
}

// ---------------------------------------------------------------------------
// Edge aggregation: one wave per edge; coalesced b128 row gather,
// global_atomic_add_f32 scatter into Agg; lane 0 counts the degree.
// ---------------------------------------------------------------------------
__global__ void edge_agg_kernel(const float* __restrict__ H,
                                const int* __restrict__ src,
                                const int* __restrict__ dst,
                                float* __restrict__ Agg,
                                float* __restrict__ Deg, int E)
{
    const int lane = threadIdx.x & 31;
    const int e = (int)((blockIdx.x * blockDim.x + threadIdx.x) >> 5);
    if (e >= E) return;

    const int s = src[e];
    const int d = dst[e];

    const float4 m = *(const float4*)(H + (size_t)s * DHID + lane * 4);
    float* out = Agg + (size_t)d * DHID + lane * 4;
    atomicAdd(out + 0, m.x);
    atomicAdd(out + 1, m.y);
    atomicAdd(out + 2, m.z);
    atomicAdd(out + 3, m.w);
    if (lane == 0) atomicAdd(Deg + d, 1.0f);
}

extern "C" void kernel_launch(void* const* d_in, const int* in_sizes, int n_in,
                              void* d_out, int out_size, void* d_ws, size_t ws_size,
                              hipStream_t stream)
{
    (void)n_in; (void)out_size; (void)ws_size;

    const float* features = (const float*)d_in[0];
    const int*   src0     = (const int*)d_in[1];
    const int*   dst0     = (const int*)d_in[2];
    const int*   src1     = (const int*)d_in[3];
    const int*   dst1     = (const int*)d_in[4];
    const float* W_init   = (const float*)d_in[5];
    const float* b_init   = (const float*)d_in[6];
    const float* W_self   = (const float*)d_in[7];
    const float* b_self   = (const float*)d_in[8];
    const float* W_neigh  = (const float*)d_in[9];
    const float* b_neigh  = (const float*)d_in[10];

    const int N_SRC0 = 200000, N_DST0 = 50000, N_DST1 = 10000;
    const int E0 = in_sizes[1];   // 800000
    const int E1 = in_sizes[3];   // 160000

    // workspace layout (floats); all offsets 16B-aligned
    float* ws   = (float*)d_ws;
    float* h0   = ws;                                 // 200000*128
    float* agg0 = h0   + (size_t)N_SRC0 * DHID;       // 50000*128
    float* deg0 = agg0 + (size_t)N_DST0 * DHID;       // 50000
    float* h1   = deg0 + N_DST0;                      // 50000*128
    float* agg1 = h1   + (size_t)N_DST0 * DHID;       // 10000*128
    float* deg1 = agg1 + (size_t)N_DST1 * DHID;       // 10000

    hipMemsetAsync(agg0, 0, ((size_t)N_DST0 * DHID + N_DST0) * sizeof(float), stream);
    hipMemsetAsync(agg1, 0, ((size_t)N_DST1 * DHID + N_DST1) * sizeof(float), stream);

    const int THREADS = 256;                 // 8 wave32s per block
    const int WPB = THREADS / 32;
    const size_t LDS_ONE = (size_t)KPAIRS * NPAD * sizeof(v2f);   // 73,728 B
    const size_t LDS_TWO = 2 * LDS_ONE;                           // 147,456 B

    // 1) h0 = relu(features @ W_init + b_init)
    {
        int strips = N_SRC0 / 16;                    // 12500
        int blocks = (strips + WPB - 1) / WPB;       // 1563
        if (blocks > 1024) blocks = 1024;            // amortize W staging
        fc_kernel<true><<<blocks, THREADS, LDS_ONE, stream>>>(
            features, W_init, b_init, h0, N_SRC0);
    }

    // 2) aggregate edges of block 0 into agg0/deg0
    {
        int blocks = (E0 + WPB - 1) / WPB;
        edge_agg_kernel<<<blocks, THREADS, 0, stream>>>(h0, src0, dst0, agg0, deg0, E0);
    }

    // 3) h1 = relu(h0[:50000] @ W_self + b_self + mean0 @ W_neigh + b_neigh)
    {
        int strips = N_DST0 / 16;                    // 3125
        int blocks = (strips + WPB - 1) / WPB;       // 391
        sage_kernel<true><<<blocks, THREADS, LDS_TWO, stream>>>(
            h0, agg0, deg0, W_self, b_self, W_neigh, b_neigh, h1, N_DST0);
    }

    // 4) aggregate edges of block 1 into agg1/deg1
    {
        int blocks = (E1 + WPB - 1) / WPB;
        edge_agg_kernel<<<blocks, THREADS, 0, stream>>>(h1, src1, dst1, agg1, deg1, E1);
    }

    // 5) out = h1[:10000] @ W_self + b_self + mean1 @ W_neigh + b_neigh  (no relu)
    {
        int strips = N_DST1 / 16;                    // 625
        int blocks = (strips + WPB - 1) / WPB;       // 79
        sage_kernel<false><<<blocks, THREADS, LDS_TWO, stream>>>(
            h1, agg1, deg1, W_self, b_self, W_neigh, b_neigh, (float*)d_out, N_DST1);
    }
}